// GraphConvolution_12756052869313
// MI455X (gfx1250) — compile-verified
//
#include <hip/hip_runtime.h>
#include <math.h>
#include <stdint.h>

typedef __attribute__((ext_vector_type(2))) float v2f;
typedef __attribute__((ext_vector_type(4))) float v4f;
typedef __attribute__((ext_vector_type(8))) float v8f;
typedef __attribute__((ext_vector_type(4))) unsigned int u32x4;
typedef __attribute__((ext_vector_type(4))) int i32x4;
typedef __attribute__((ext_vector_type(8))) int i32x8;

#define BDIM   16
#define NDIM   1024
#define FDIM   1024
#define ODIM   1024
#define MTILE  32                       // rows of the output stripe per workgroup
#define WAVES  8
#define TPB    (WAVES * 32)             // 256 threads = 8 wave32
#define RS_A   1032                     // sA row stride: TDM pad = 2 dwords per 256 dwords
#define LDW    (FDIM + 4)               // sS row stride (manual pad, conflict-free)
#define TILES_PER_WAVE (FDIM / 16 / WAVES)  // 8 column tiles of 16 per wave
#define SA_FLOATS (MTILE * RS_A)        // 33024 floats (includes trailing pad)
#define SS_FLOATS (MTILE * LDW)
#define SMEM_BYTES ((SA_FLOATS + SS_FLOATS) * 4)   // 263,680 B (< 320KB WGP LDS)

// sA element (r,c): TDM inserts 2 pad dwords after every 256 stored dwords.
// linear index = r*1024 + c ; pads before it = 2*floor((r*1024+c)/256) = 8r + 2*(c>>8)
__device__ __forceinline__ int aidx(int r, int c) {
    return r * RS_A + c + 2 * (c >> 8);
}

// Fused GCNII layer:
//   S  = (1-alpha) * (adj[row0:row0+32,:] @ X_b) + alpha * H0_b[row0:row0+32,:]
//   out = theta * (S @ W) + (1-theta) * S
// adj stripe arrives in LDS via the Tensor Data Mover (hardware-padded rows);
// S lives entirely in LDS between the two GEMMs (no global intermediate).
__launch_bounds__(TPB)
__global__ void gcnii_fused_wmma(const float* __restrict__ x,
                                 const float* __restrict__ adj,
                                 const float* __restrict__ h0,
                                 const float* __restrict__ w,
                                 const float* __restrict__ p_lamda,
                                 const float* __restrict__ p_alpha,
                                 const int*   __restrict__ p_l,
                                 float* __restrict__ out)
{
    extern __shared__ float smem[];
    float* sA = smem;                    // adj stripe    [MTILE][RS_A] (TDM-padded)
    float* sS = smem + SA_FLOATS;        // support stripe[MTILE][LDW]

    const int b    = blockIdx.y;
    const int row0 = blockIdx.x * MTILE;
    const int tid  = threadIdx.x;
    const int wave = tid >> 5;
    const int lane = tid & 31;
    const int l15  = lane & 15;          // M (A-frag / C rows) or N (B-frag / C cols)
    const int hi   = lane >> 4;          // K half-select per 16x16x4 fp32 layout

    // scalars come from device memory (graph-capture safe)
    const float lamda = p_lamda[0];
    const float alpha = p_alpha[0];
    const float lnum  = (float)p_l[0];
    const float theta = fminf(1.0f, logf(lamda / lnum + 1.0f));

    // ---- TDM: DMA adj[row0:row0+32, 0:1024] (fp32) into sA with HW row padding ----
#if __has_builtin(__builtin_amdgcn_tensor_load_to_lds)
    if (tid < 32) {                      // one wave issues the DMA (TDM ignores EXEC)
        const uint64_t ga = (uint64_t)(uintptr_t)(adj + (size_t)row0 * NDIM);
        const uint32_t lds_off = (uint32_t)(uintptr_t)(void*)sA; // flat->LDS offset (low 32b)
        // D# group 0: count=1 | lds_addr | global_addr[56:0] | type=2 ("image")
        const u32x4 g0 = { 1u,
                           lds_off,
                           (uint32_t)ga,
                           (uint32_t)((ga >> 32) & 0x01FFFFFFu) | (2u << 30) };
        // D# group 1:
        //  w0: wg_mask=0 | data_size=2 (4B) | pad_enable | pad_interval=7 (256dw) | pad_amount=1 (2dw)
        //  w1[63:48]=tensor_dim0.lo=1024   w2[95:80]=tensor_dim1.lo=1024
        //  w3[127:112]=tile_dim0=1024      w4[143:128]=tile_dim1=32 (tile_dim2=0)
        //  w5=tensor_dim0_stride=1024      w6,w7=0 (2D)
        const i32x8 g1 = { (int)0x03D20000,
                           (int)(1024u << 16),
                           (int)(1024u << 16),
                           (int)(1024u << 16),
                           (int)MTILE,
                           (int)1024,
                           0, 0 };
        const i32x4 gz = { 0, 0, 0, 0 };
#if __clang_major__ >= 23
        const i32x8 gz8 = { 0, 0, 0, 0, 0, 0, 0, 0 };
        __builtin_amdgcn_tensor_load_to_lds(g0, g1, gz, gz, gz8, 0);
#else
        __builtin_amdgcn_tensor_load_to_lds(g0, g1, gz, gz, 0);
#endif
        __builtin_amdgcn_s_wait_tensorcnt(0);
    }
#else
    // Fallback: cooperative staging into the same padded layout
    for (int i = tid; i < MTILE * (NDIM / 4); i += TPB) {
        const int r = i / (NDIM / 4);
        const int c = (i - r * (NDIM / 4)) * 4;
        v4f v = *(const v4f*)(adj + (size_t)(row0 + r) * NDIM + c);
        v2f lo = { v.x, v.y }, hiv = { v.z, v.w };
        *(v2f*)(&sA[aidx(r, c)    ]) = lo;   // quad never straddles a pad (c % 4 == 0)
        *(v2f*)(&sA[aidx(r, c) + 2]) = hiv;
    }
#endif
    __syncthreads();

    const float* Xb  = x  + (size_t)b * NDIM * FDIM;
    const float* H0b = h0 + (size_t)b * NDIM * FDIM;

    // ---- Stage A: S = (1-alpha) * adj_stripe @ X_b + alpha * H0 stripe ----
    for (int t = 0; t < TILES_PER_WAVE; ++t) {
        const int n0 = (wave * TILES_PER_WAVE + t) * 16;
        v8f acc0 = {};                   // rows row0 +  0..15
        v8f acc1 = {};                   // rows row0 + 16..31
        #pragma unroll 4
        for (int k = 0; k < NDIM; k += 4) {
            // B 4x16 fp32: K striped across lane halves, N = lane&15
            v2f bfrag;
            bfrag.x = Xb[(size_t)(k + 2 * hi)     * FDIM + n0 + l15];
            bfrag.y = Xb[(size_t)(k + 2 * hi + 1) * FDIM + n0 + l15];
            // A 16x4 fp32: lane<16 -> M=lane, K=k..k+1 ; lane>=16 -> K=k+2..k+3
            v2f a0 = *(const v2f*)(&sA[aidx(l15,      k + 2 * hi)]);
            v2f a1 = *(const v2f*)(&sA[aidx(l15 + 16, k + 2 * hi)]);
            acc0 = __builtin_amdgcn_wmma_f32_16x16x4_f32(
                false, a0, false, bfrag, (short)0, acc0, false, false);
            acc1 = __builtin_amdgcn_wmma_f32_16x16x4_f32(
                false, a1, false, bfrag, (short)0, acc1, false, false);
        }
        // C/D layout: VGPR r -> M = r + 8*hi, N = lane&15
        #pragma unroll
        for (int r = 0; r < 8; ++r) {
            const int m = r + 8 * hi;
            const float sv0 = (1.0f - alpha) * acc0[r]
                            + alpha * H0b[(size_t)(row0 + m) * FDIM + n0 + l15];
            const float sv1 = (1.0f - alpha) * acc1[r]
                            + alpha * H0b[(size_t)(row0 + 16 + m) * FDIM + n0 + l15];
            sS[(m     ) * LDW + n0 + l15] = sv0;
            sS[(m + 16) * LDW + n0 + l15] = sv1;
        }
    }
    __syncthreads();

    // ---- Stage B: out = theta * (S @ W) + (1-theta) * S ----
    for (int t = 0; t < TILES_PER_WAVE; ++t) {
        const int o0 = (wave * TILES_PER_WAVE + t) * 16;
        v8f acc0 = {};
        v8f acc1 = {};
        #pragma unroll 4
        for (int k = 0; k < FDIM; k += 4) {
            v2f bfrag;
            bfrag.x = w[(size_t)(k + 2 * hi)     * ODIM + o0 + l15];
            bfrag.y = w[(size_t)(k + 2 * hi + 1) * ODIM + o0 + l15];
            v2f a0 = *(const v2f*)(&sS[(l15     ) * LDW + k + 2 * hi]);
            v2f a1 = *(const v2f*)(&sS[(l15 + 16) * LDW + k + 2 * hi]);
            acc0 = __builtin_amdgcn_wmma_f32_16x16x4_f32(
                false, a0, false, bfrag, (short)0, acc0, false, false);
            acc1 = __builtin_amdgcn_wmma_f32_16x16x4_f32(
                false, a1, false, bfrag, (short)0, acc1, false, false);
        }
        #pragma unroll
        for (int r = 0; r < 8; ++r) {
            const int m = r + 8 * hi;
            out[((size_t)b * NDIM + row0 + m) * ODIM + o0 + l15] =
                theta * acc0[r] + (1.0f - theta) * sS[(m     ) * LDW + o0 + l15];
            out[((size_t)b * NDIM + row0 + 16 + m) * ODIM + o0 + l15] =
                theta * acc1[r] + (1.0f - theta) * sS[(m + 16) * LDW + o0 + l15];
        }
    }
}

extern "C" void kernel_launch(void* const* d_in, const int* in_sizes, int n_in,
                              void* d_out, int out_size, void* d_ws, size_t ws_size,
                              hipStream_t stream) {
    (void)in_sizes; (void)n_in; (void)out_size; (void)d_ws; (void)ws_size;
    const float* x    = (const float*)d_in[0];   // prott5_emd [B,N,F]
    const float* adj  = (const float*)d_in[1];   // [N,N]
    const float* h0   = (const float*)d_in[2];   // [B,N,F]
    const float* w    = (const float*)d_in[3];   // [F,OUT]
    const float* lam  = (const float*)d_in[4];   // scalar
    const float* alph = (const float*)d_in[5];   // scalar
    const int*   lv   = (const int*)d_in[6];     // scalar (int)
    float* out = (float*)d_out;                  // [B,N,OUT]

    // Allow >64KB dynamic LDS (CDNA5 WGP: 320KB). Not a stream op -> capture-safe.
    (void)hipFuncSetAttribute((const void*)gcnii_fused_wmma,
                              hipFuncAttributeMaxDynamicSharedMemorySize,
                              SMEM_BYTES);

    dim3 grid(NDIM / MTILE, BDIM);               // 32 x 16 workgroups
    gcnii_fused_wmma<<<grid, TPB, SMEM_BYTES, stream>>>(x, adj, h0, w, lam, alph, lv, out);
}